// DifferentiableEKF_50122268344600
// MI455X (gfx1250) — compile-verified
//
#include <hip/hip_runtime.h>

// ---------------------------------------------------------------------------
// Differentiable EKF scan on MI455X (gfx1250), wave32 + V_WMMA_F32_16X16X4_F32
// One wave per batch element; T-loop sequential inside the wave.
// Round 3: branch-free Gauss-Jordan (full-snapshot b128 broadcast, cndmask
// selects instead of EXEC-masked stores) to kill the round-2 SALU blowup.
// ---------------------------------------------------------------------------

typedef __attribute__((ext_vector_type(2))) float v2f;
typedef __attribute__((ext_vector_type(8))) float v8f;

#define WPB 4  // waves per block

// Compiler-only memory fence: same-wave LDS ops are serviced in order by HW
// (DScnt in-order per wave); this just stops clang reordering cross-lane deps.
__device__ __forceinline__ void wsync() { asm volatile("" ::: "memory"); }

// ---- WMMA f32 16x16x4 fragment addressing (row-major 16x16 LDS tiles) ----
// A-matrix 16x4 (MxK): lanes 0-15 = M rows, K=0,1 in v[0],v[1]; lanes 16-31
// hold K=2,3.  B-matrix 4x16 (KxN): lane = N column, VGPR = K row pair.
__device__ __forceinline__ v2f afrag(const float* M, int kk, int lane) {
  int m = lane & 15, kh = lane >> 4;
  int k0 = 4 * kk + 2 * kh;
  v2f a;
  a.x = M[m * 16 + k0];
  a.y = M[m * 16 + k0 + 1];
  return a;
}
__device__ __forceinline__ v2f bfrag(const float* M, int kk, int lane) {
  int n = lane & 15, kh = lane >> 4;
  int k0 = 4 * kk + 2 * kh;
  v2f b;
  b.x = M[k0 * 16 + n];
  b.y = M[(k0 + 1) * 16 + n];
  return b;
}
__device__ __forceinline__ v2f bfragT(const float* M, int kk, int lane) {
  int n = lane & 15, kh = lane >> 4;
  int k0 = 4 * kk + 2 * kh;
  v2f b;
  b.x = M[n * 16 + k0];
  b.y = M[n * 16 + k0 + 1];
  return b;
}

struct Frag {
  v2f k[4];
};
__device__ __forceinline__ Frag loadA(const float* M, int lane) {
  Frag f;
#pragma unroll
  for (int kk = 0; kk < 4; ++kk) f.k[kk] = afrag(M, kk, lane);
  return f;
}
__device__ __forceinline__ Frag loadB(const float* M, int lane) {
  Frag f;
#pragma unroll
  for (int kk = 0; kk < 4; ++kk) f.k[kk] = bfrag(M, kk, lane);
  return f;
}
__device__ __forceinline__ Frag loadBT(const float* M, int lane) {
  Frag f;
#pragma unroll
  for (int kk = 0; kk < 4; ++kk) f.k[kk] = bfragT(M, kk, lane);
  return f;
}

// C/D 16x16 f32 layout: VGPR r, lanes 0-15 -> row r, lanes 16-31 -> row r+8.
__device__ __forceinline__ v8f loadC16(const float* M, int lane) {
  int col = lane & 15, hi = lane >> 4;
  v8f c;
#pragma unroll
  for (int r = 0; r < 8; ++r) c[r] = M[(r + 8 * hi) * 16 + col];
  return c;
}
__device__ __forceinline__ void storeD16(float* M, v8f d, int lane) {
  int col = lane & 15, hi = lane >> 4;
#pragma unroll
  for (int r = 0; r < 8; ++r) M[(r + 8 * hi) * 16 + col] = d[r];
}

// 16x16x16 f32 matmul = 4 chained v_wmma_f32_16x16x4_f32 on fragments.
__device__ __forceinline__ v8f mmrr(const Frag& a, const Frag& b, v8f acc) {
#pragma unroll
  for (int kk = 0; kk < 4; ++kk) {
    acc = __builtin_amdgcn_wmma_f32_16x16x4_f32(false, a.k[kk], false, b.k[kk],
                                                (short)0, acc, false, false);
  }
  return acc;
}

// 0.5*(P + P^T) + eps*I ; reads the transposed element of the LDS copy.
__device__ __forceinline__ v8f symeps(const float* M, v8f d, int lane, float eps) {
  int col = lane & 15, hi = lane >> 4;
#pragma unroll
  for (int r = 0; r < 8; ++r) {
    int row = r + 8 * hi;
    float t = M[col * 16 + row];
    d[r] = 0.5f * (d[r] + t) + ((row == col) ? eps : 0.0f);
  }
  return d;
}

// ---------------------------------------------------------------------------
// Prologue: Q = Lq Lq^T (16x16), R = Lr Lr^T (8x8, zero-padded to 16x16) -> ws
// ---------------------------------------------------------------------------
__global__ __launch_bounds__(256) void ekf_prep(const float* __restrict__ ldQ,
                                                const float* __restrict__ odQ,
                                                const float* __restrict__ ldR,
                                                const float* __restrict__ odR,
                                                float* __restrict__ ws) {
  __shared__ float Lq[256];
  __shared__ float Lr[64];
  int t = threadIdx.x;
  {
    int r = t >> 4, c = t & 15;
    float v = 0.0f;
    if (r == c)
      v = fmaxf(__expf(ldQ[r]), 1e-6f);
    else if (r > c)
      v = odQ[r * (r - 1) / 2 + c];  // np.tril_indices(n,-1) row-major order
    Lq[t] = v;
  }
  if (t < 64) {
    int r = t >> 3, c = t & 7;
    float v = 0.0f;
    if (r == c)
      v = fmaxf(__expf(ldR[r]), 1e-6f);
    else if (r > c)
      v = odR[r * (r - 1) / 2 + c];
    Lr[t] = v;
  }
  __syncthreads();
  {
    int r = t >> 4, c = t & 15;
    float s = 0.0f;
#pragma unroll
    for (int k = 0; k < 16; ++k) s += Lq[r * 16 + k] * Lq[c * 16 + k];
    ws[t] = s;  // Q16
    float sr = 0.0f;
    if (r < 8 && c < 8) {
#pragma unroll
      for (int k = 0; k < 8; ++k) sr += Lr[r * 8 + k] * Lr[c * 8 + k];
    }
    ws[256 + t] = sr;  // R padded to 16x16
  }
}

// ---------------------------------------------------------------------------
// Main scan kernel: one wave32 per batch element, 4 waves per block.
// ---------------------------------------------------------------------------
__global__ __launch_bounds__(32 * WPB) void ekf_scan(
    const float* __restrict__ x0, const float* __restrict__ P0,
    const float* __restrict__ meas, const float* __restrict__ Amat,
    const float* __restrict__ Cmat, const float* __restrict__ QR,
    float* __restrict__ out, int B, int T) {
  __shared__ __align__(16) float cA[256];
  __shared__ __align__(16) float cC[256];
  __shared__ __align__(16) float cQ[256];
  __shared__ __align__(16) float cR[256];
  __shared__ __align__(16) float sP[WPB][256];
  __shared__ __align__(16) float sT[WPB][256];
  __shared__ __align__(16) float sK[WPB][256];
  __shared__ __align__(16) float sIK[WPB][256];
  __shared__ __align__(16) float sSI[WPB][256];
  __shared__ __align__(16) float sAUG[WPB][128];  // GJ snapshot [S | I] 8x16
  __shared__ float sX[WPB][16], sXP[WPB][16], sInn[WPB][8];

  const int tid = threadIdx.x;
  const int w = tid >> 5, lane = tid & 31;
  const int b = blockIdx.x * WPB + w;

  for (int i = tid; i < 256; i += 32 * WPB) {
    cA[i] = Amat[i];
    cQ[i] = QR[i];
    cR[i] = QR[256 + i];
    cC[i] = (i < 128) ? Cmat[i] : 0.0f;  // C (8x16) zero-padded to 16x16
  }
  __syncthreads();

  float* P = sP[w];
  float* Tm = sT[w];
  float* K = sK[w];
  float* IK = sIK[w];
  float* SI = sSI[w];
  float* AUG = sAUG[w];

  // ---- hoist loop-invariant fragments into registers ----
  const Frag fA = loadA(cA, lane);     // A   as A-operand
  const Frag fATb = loadBT(cA, lane);  // A^T as B-operand
  const Frag fCa = loadA(cC, lane);    // C   as A-operand
  const Frag fCb = loadB(cC, lane);    // C   as B-operand
  const Frag fCTb = loadBT(cC, lane);  // C^T as B-operand
  const Frag fRb = loadB(cR, lane);    // R   as B-operand
  const v8f accQ = loadC16(cQ, lane);  // Q   as C accumulator
  const v8f accR = loadC16(cR, lane);  // R   as C accumulator
  // lanes 0-15: row `lane` of A;  lanes 16-23: row `lane-16` of C (matvecs)
  float arow[16];
#pragma unroll
  for (int j = 0; j < 16; ++j) {
    float v = 0.0f;
    if (lane < 16)
      v = cA[lane * 16 + j];
    else if (lane < 24)
      v = cC[(lane - 16) * 16 + j];
    arow[j] = v;
  }

  // ---- initial state / covariance ----
  {
    const float4* src = (const float4*)(P0 + (size_t)b * 256);
    float4* dstp = (float4*)P;
    dstp[lane * 2] = src[lane * 2];
    dstp[lane * 2 + 1] = src[lane * 2 + 1];
    if (lane < 16) sX[w][lane] = x0[(size_t)b * 16 + lane];
  }
  wsync();

  const size_t off_cov = (size_t)B * T * 16;
  const size_t off_z = off_cov + (size_t)B * T * 256;
  const float EPSI = 1e-6f;

  v8f z;
#pragma unroll
  for (int r = 0; r < 8; ++r) z[r] = 0.0f;

  const int gr = lane >> 2;        // GJ: owned row 0..7
  const int gc0 = (lane & 3) * 4;  // GJ: first owned column (0,4,8,12)

  for (int t = 0; t < T; ++t) {
    // ---- predict mean: x_pred = A x (lanes 0-15) ----
    if (lane < 16) {
      float s = 0.0f;
#pragma unroll
      for (int j = 0; j < 16; ++j) s += arow[j] * sX[w][j];
      sXP[w][lane] = s;
    }
    wsync();
    // ---- z_pred = C x_pred ; innovation (lanes 16-23) ----
    if (lane >= 16 && lane < 24) {
      int m = lane - 16;
      float s = 0.0f;
#pragma unroll
      for (int j = 0; j < 16; ++j) s += arow[j] * sXP[w][j];
      out[off_z + ((size_t)b * T + t) * 8 + m] = s;
      sInn[w][m] = meas[((size_t)b * T + t) * 8 + m] - s;
    }
    wsync();

    // ---- P_pred = A P A^T + Q, regularized ----
    v8f t1 = mmrr(fA, loadB(P, lane), z);  // A @ P
    storeD16(Tm, t1, lane);
    wsync();
    v8f pp = mmrr(loadA(Tm, lane), fATb, accQ);  // (AP) @ A^T + Q
    storeD16(P, pp, lane);
    wsync();
    pp = symeps(P, pp, lane, EPSI);
    wsync();
    storeD16(P, pp, lane);
    wsync();

    // ---- S = C P_pred C^T + R, regularized ----
    v8f cp = mmrr(fCa, loadB(P, lane), z);  // C @ P_pred
    storeD16(Tm, cp, lane);
    wsync();
    v8f sm = mmrr(loadA(Tm, lane), fCTb, accR);  // @ C^T + R
    storeD16(SI, sm, lane);
    wsync();
    sm = symeps(SI, sm, lane, EPSI);
    wsync();
    storeD16(SI, sm, lane);
    wsync();

    // ---- invert 8x8 S: branch-free register Gauss-Jordan on [S | I] ----
    // lane owns row gr, columns gc0..gc0+3.  Each pivot step: one b128
    // snapshot store, three loads (pivot, row factor, pivot-row chunk),
    // then pure VALU updates with cndmask selects (no EXEC masking).
    float av[4];
    {
      float4 s4 = *(const float4*)&SI[gr * 16 + gc0];
      const float sv[4] = {s4.x, s4.y, s4.z, s4.w};
#pragma unroll
      for (int i = 0; i < 4; ++i) {
        int c = gc0 + i;
        av[i] = (c < 8) ? sv[i] : ((c - 8 == gr) ? 1.0f : 0.0f);
      }
    }
#pragma unroll
    for (int k = 0; k < 8; ++k) {
      *(float4*)&AUG[lane * 4] = make_float4(av[0], av[1], av[2], av[3]);
      wsync();
      float pinv = 1.0f / AUG[k * 16 + k];       // pivot (uniform)
      float fr = AUG[gr * 16 + k];               // my row's factor
      float4 rk = *(const float4*)&AUG[k * 16 + gc0];  // pivot-row chunk
      wsync();
      const float rb[4] = {rk.x, rk.y, rk.z, rk.w};
#pragma unroll
      for (int i = 0; i < 4; ++i) {
        float scaled = rb[i] * pinv;
        av[i] = (gr == k) ? scaled : (av[i] - fr * scaled);
      }
    }
    // SI <- S^{-1} zero-padded to 16x16 (inverse sits in columns 8..15)
    {
      float4 z4 = make_float4(0.0f, 0.0f, 0.0f, 0.0f);
      float4* sip = (float4*)SI;
      sip[lane * 2] = z4;
      sip[lane * 2 + 1] = z4;
    }
    wsync();
    if (gc0 >= 8)
      *(float4*)&SI[gr * 16 + (gc0 - 8)] = make_float4(av[0], av[1], av[2], av[3]);
    wsync();

    // ---- K = P_pred C^T S^{-1} ----
    v8f pht = mmrr(loadA(P, lane), fCTb, z);  // P_pred @ C^T (cols 8-15 zero)
    storeD16(Tm, pht, lane);
    wsync();
    v8f kf = mmrr(loadA(Tm, lane), loadB(SI, lane), z);  // @ S^{-1}
    storeD16(K, kf, lane);
    wsync();

    // ---- x_new = x_pred + K innov ----
    if (lane < 16) {
      float s = sXP[w][lane];
#pragma unroll
      for (int m = 0; m < 8; ++m) s += K[lane * 16 + m] * sInn[w][m];
      sX[w][lane] = s;
      out[((size_t)b * T + t) * 16 + lane] = s;
    }
    wsync();

    // ---- IKH = I - K C ----
    v8f kc = mmrr(loadA(K, lane), fCb, z);
    {
      int col = lane & 15, hi = lane >> 4;
#pragma unroll
      for (int r = 0; r < 8; ++r) {
        int row = r + 8 * hi;
        kc[r] = ((row == col) ? 1.0f : 0.0f) - kc[r];
      }
    }
    storeD16(IK, kc, lane);
    wsync();

    // ---- P_new = IKH P_pred IKH^T + K R K^T (Joseph form), regularized ----
    v8f t2 = mmrr(loadA(IK, lane), loadB(P, lane), z);  // IKH @ P_pred
    storeD16(Tm, t2, lane);
    wsync();
    v8f pn = mmrr(loadA(Tm, lane), loadBT(IK, lane), z);  // @ IKH^T
    wsync();
    v8f kr = mmrr(loadA(K, lane), fRb, z);  // K @ R
    storeD16(Tm, kr, lane);
    wsync();
    pn = mmrr(loadA(Tm, lane), loadBT(K, lane), pn);  // + (KR) @ K^T
    storeD16(P, pn, lane);
    wsync();
    pn = symeps(P, pn, lane, EPSI);
    wsync();
    storeD16(P, pn, lane);
    wsync();

    // ---- write covariance output: 2 x b128 per lane (1KB contiguous/wave) --
    {
      const float4* srcp = (const float4*)P;
      float4 q0 = srcp[lane * 2];
      float4 q1 = srcp[lane * 2 + 1];
      float4* dst = (float4*)(out + off_cov + ((size_t)b * T + t) * 256);
      dst[lane * 2] = q0;
      dst[lane * 2 + 1] = q1;
    }
  }
}

// ---------------------------------------------------------------------------
extern "C" void kernel_launch(void* const* d_in, const int* in_sizes, int n_in,
                              void* d_out, int out_size, void* d_ws, size_t ws_size,
                              hipStream_t stream) {
  const float* x0 = (const float*)d_in[0];
  const float* P0 = (const float*)d_in[1];
  const float* meas = (const float*)d_in[2];
  const float* A = (const float*)d_in[3];
  const float* C = (const float*)d_in[4];
  const float* ldQ = (const float*)d_in[5];
  const float* odQ = (const float*)d_in[6];
  const float* ldR = (const float*)d_in[7];
  const float* odR = (const float*)d_in[8];
  float* out = (float*)d_out;
  float* ws = (float*)d_ws;

  const int B = in_sizes[0] / 16;       // 2048
  const int T = in_sizes[2] / (B * 8);  // 200

  ekf_prep<<<1, 256, 0, stream>>>(ldQ, odQ, ldR, odR, ws);
  ekf_scan<<<B / WPB, 32 * WPB, 0, stream>>>(x0, P0, meas, A, C, ws, out, B, T);
}